// idxLayer_31980326486772
// MI455X (gfx1250) — compile-verified
//
#include <hip/hip_runtime.h>

// Problem constants from the reference: N=100000 rows, W=32 neighbors, D=32 feat.
#define WN 32
#define DN 32
#define ROW_OUT (WN * DN + 2 * WN)   // 1088 floats per output row

typedef float float4v __attribute__((ext_vector_type(4)));

// One wave32 per output row.
//  - lanes form 4 neighbor-groups (group = lane>>3) x 8 quad-slots (gl = lane&7)
//  - 8 iterations cover all 32 neighbors; per iteration the wave emits one
//    contiguous 512B NT store burst and 4x128B coalesced loads from x.
__global__ __launch_bounds__(256) void gather_concat_rows(
    const float* __restrict__ x,          // [N, 32] f32  (12.8 MB, L2-resident)
    const long long* __restrict__ idx,    // [N, 32] i64  (values < N, low dword valid)
    const float* __restrict__ dis,        // [N, 32] f32
    const float* __restrict__ angle,      // [N, 32] f32
    float* __restrict__ out,              // [N, 1088] f32
    int nrows)
{
    const int lane = threadIdx.x & 31;
    const int wave = threadIdx.x >> 5;             // 0..7 waves per block
    const int h    = blockIdx.x * 8 + wave;        // output row
    if (h >= nrows) return;

    const int group = lane >> 3;                   // which neighbor in quad (0..3)
    const int gl    = lane & 7;                    // which float4 of the 32-f row (0..7)

    // int64 indices, little-endian: low dword at even int32 offsets.
    const int* __restrict__ idx32 = (const int*)idx;
    const size_t idx_row = (size_t)h * WN;

    float* __restrict__ orow = out + (size_t)h * ROW_OUT;

    #pragma unroll
    for (int it = 0; it < 8; ++it) {
        const int w   = it * 4 + group;                            // neighbor id
        const int row = idx32[(idx_row + (size_t)w) * 2];          // x row to gather
        // 128B-aligned, fully coalesced per 8-lane group; RT hint -> cached in L2.
        const float4v v = *(const float4v*)(x + (size_t)row * DN + gl * 4);
        // Wave-contiguous 512B burst; NT hint keeps the 435MB write stream
        // from evicting the 12.8MB x working set out of the 192MB L2.
        __builtin_nontemporal_store(v, (float4v*)(orow + w * DN + gl * 4));
    }

    // Tail: dis (cols 1024..1055) and angle (cols 1056..1087), 1 float/lane,
    // each a 128B coalesced NT transaction (streamed once, never reused).
    const float dv = __builtin_nontemporal_load(dis   + idx_row + lane);
    const float av = __builtin_nontemporal_load(angle + idx_row + lane);
    __builtin_nontemporal_store(dv, orow + WN * DN + lane);
    __builtin_nontemporal_store(av, orow + WN * DN + WN + lane);
}

extern "C" void kernel_launch(void* const* d_in, const int* in_sizes, int n_in,
                              void* d_out, int out_size, void* d_ws, size_t ws_size,
                              hipStream_t stream) {
    const float*     x     = (const float*)d_in[0];
    const long long* idx   = (const long long*)d_in[1];
    const float*     dis   = (const float*)d_in[2];
    const float*     angle = (const float*)d_in[3];
    float*           out   = (float*)d_out;

    const int nrows = in_sizes[0] / DN;            // N = 100000
    const int blocks = (nrows + 7) / 8;            // 8 rows (waves) per 256-thr block

    gather_concat_rows<<<dim3(blocks), dim3(256), 0, stream>>>(
        x, idx, dis, angle, out, nrows);
}